// VN_DGCNN_Encoder_24687472017761
// MI455X (gfx1250) — compile-verified
//
#include <hip/hip_runtime.h>
#include <math.h>

// ---------------------------------------------------------------------------
// VN-DGCNN encoder for gfx1250 (MI455X), fp32 WMMA (V_WMMA_F32_16X16X4_F32).
// Working set (~41MB) is L2-resident (192MB) -> matrix-pipe bound.
// All hot-loop memory ops are branchless (no EXEC-mask juggling): block-1
// operands are zero-padded up front so every K-loop has C,ci % 4 == 0.
// ---------------------------------------------------------------------------

typedef __attribute__((ext_vector_type(2))) float v2f;
typedef __attribute__((ext_vector_type(8))) float v8f;

#define NPTS 1024
#define KNN 20
#define EPSV 1e-6f
#define BNEPS 1e-5f

__device__ __forceinline__ v8f wmma4(v2f a, v2f b, v8f c) {
  // D = A(16x4 f32) * B(4x16 f32) + C(16x16 f32)
  return __builtin_amdgcn_wmma_f32_16x16x4_f32(false, a, false, b, (short)0, c,
                                               false, false);
}

// Edge feature, fully branchless (both loads always issue, value selected by
// v_cndmask). Row index clamped into [0, Cin) — out-of-range channels only
// occur for block 1 where the padded weights are zero, so the (finite)
// clamped value contributes nothing.
__device__ __forceinline__ float gfeat(const float* __restrict__ Xb, int i,
                                       int d, int n, int nbr, int c, int Cin) {
  int row = ((i < c) ? i : (i - c)) * 3 + d;
  row = (row < Cin) ? row : 0;
  const float* r = Xb + row * NPTS;
  const float cen = r[n];
  const float nb = r[nbr];
  return (i < c) ? (nb - cen) : cen;
}

// ---------------------------------------------------------------------------
// Prep: pad x (4,3,1024) -> xpad (4,4,1024) with zero 4th row.
// ---------------------------------------------------------------------------
__global__ __launch_bounds__(256) void padx_kernel(const float* __restrict__ x,
                                                   float* __restrict__ xpad) {
  const int t = blockIdx.x * 256 + threadIdx.x;  // 4*4*1024
  const int b = t >> 12, r = (t >> 10) & 3, n = t & (NPTS - 1);
  xpad[t] = (r < 3) ? x[(b * 3 + r) * NPTS + n] : 0.f;
}

// Prep: pad W (64,2) -> (64,4) zero-padded columns.
__global__ __launch_bounds__(256) void padw_kernel(const float* __restrict__ w,
                                                   float* __restrict__ wp) {
  const int t = threadIdx.x;  // 256 = 64*4
  const int o = t >> 2, i = t & 3;
  wp[t] = (i < 2) ? w[o * 2 + i] : 0.f;
}

// ---------------------------------------------------------------------------
// xx[b][n] = sum_c X[c,n]^2   (C % 4 == 0, padded rows are zero)
// ---------------------------------------------------------------------------
__global__ __launch_bounds__(256) void sqnorm_kernel(
    const float* __restrict__ X, int bstride, int C, float* __restrict__ xx) {
  const int t = blockIdx.x * 256 + threadIdx.x;  // b*N + n
  const int b = t >> 10, n = t & (NPTS - 1);
  const float* Xb = X + b * bstride;
  float s = 0.f;
  for (int cc = 0; cc < C; ++cc) {
    const float v = Xb[cc * NPTS + n];
    s += v * v;
  }
  xx[t] = s;
}

// ---------------------------------------------------------------------------
// dist[b][i][j] = 2*<x_i,x_j> - |x_i|^2 - |x_j|^2 via WMMA Gram tiles.
// One wave per 16x16 tile; no guards (C % 4 == 0).
// ---------------------------------------------------------------------------
__global__ __launch_bounds__(32) void gram_dist_kernel(
    const float* __restrict__ X, int bstride, int C,
    const float* __restrict__ xx, float* __restrict__ dist) {
  const int lane = threadIdx.x, half = lane >> 4, lcol = lane & 15;
  const int j0 = blockIdx.x * 16, i0 = blockIdx.y * 16, b = blockIdx.z;
  const float* Xb = X + b * bstride;
  v8f acc = {};
#pragma unroll 2
  for (int k0 = 0; k0 < C; k0 += 4) {
    const int ka = k0 + half * 2;  // A 16x4: lanes 0-15 K={0,1}, 16-31 K={2,3}
    v2f a, bm;
    a.x = Xb[ka * NPTS + i0 + lcol];
    a.y = Xb[(ka + 1) * NPTS + i0 + lcol];
    bm.x = Xb[ka * NPTS + j0 + lcol];
    bm.y = Xb[(ka + 1) * NPTS + j0 + lcol];
    acc = wmma4(a, bm, acc);
  }
  const float* xxb = xx + b * NPTS;
  const float xj = xxb[j0 + lcol];
#pragma unroll
  for (int v = 0; v < 8; ++v) {  // C/D layout: row = v + half*8, col = lcol
    const int i = i0 + v + half * 8;
    dist[(b << 20) + (i << 10) + (j0 + lcol)] = 2.f * acc[v] - xxb[i] - xj;
  }
}

// ---------------------------------------------------------------------------
// Top-20 (descending, low-index tie-break) per distance row.
// One wave per row; 32 values/lane in registers + taken-bitmask.
// ---------------------------------------------------------------------------
__global__ __launch_bounds__(256) void topk_kernel(
    const float* __restrict__ dist, int* __restrict__ outIdx) {
  const int wid = (blockIdx.x * blockDim.x + threadIdx.x) >> 5;  // row = b*N+n
  const int lane = threadIdx.x & 31;
  const float* row = dist + ((size_t)wid << 10);
  float v[32];
#pragma unroll
  for (int j = 0; j < 32; ++j) v[j] = row[j * 32 + lane];  // col = j*32+lane
  unsigned taken = 0u;
  int* orow = outIdx + wid * KNN;
  for (int r = 0; r < KNN; ++r) {
    float best = -3.4e38f;
    int bj = 0;
#pragma unroll
    for (int j = 0; j < 32; ++j) {
      const bool ok = ((taken >> j) & 1u) == 0u;
      if (ok && v[j] > best) { best = v[j]; bj = j; }
    }
    int col = bj * 32 + lane;
#pragma unroll
    for (int off = 16; off >= 1; off >>= 1) {
      const float ov = __shfl_xor(best, off);
      const int oc = __shfl_xor(col, off);
      if (ov > best || (ov == best && oc < col)) { best = ov; col = oc; }
    }
    if ((col & 31) == lane) taken |= 1u << (col >> 5);
    if (lane == 0) orow[r] = col;
  }
}

// ---------------------------------------------------------------------------
// Pass A: p = Wf x F, accumulate per-channel sum(norm)/sum(norm^2) for BN.
// GATHER=true: edge-feature gather fused into the B-tile load (blocks 1-4).
// GATHER=false: plain channel load (block 5).
// ---------------------------------------------------------------------------
template <bool GATHER>
__global__ __launch_bounds__(32) void vn_passA(
    const float* __restrict__ Xin, int bstride, int Cin,
    const float* __restrict__ Wf, const int* __restrict__ nidx, int c, int ci,
    int knn, float* __restrict__ sumN, float* __restrict__ sumN2) {
  const int lane = threadIdx.x, half = lane >> 4, lcol = lane & 15;
  const int n0 = blockIdx.x * 16, o0 = blockIdx.y * 16, b = blockIdx.z;
  const float* Xb = Xin + b * bstride;
  const int ncol = n0 + lcol;
  const float* wfrow = Wf + (o0 + lcol) * ci;
  const int* irow = nidx + (b * NPTS + ncol) * KNN;
  float pn[8], pn2[8];
#pragma unroll
  for (int v = 0; v < 8; ++v) { pn[v] = 0.f; pn2[v] = 0.f; }

  for (int k = 0; k < knn; ++k) {
    const int nbr = GATHER ? irow[k] : 0;
    v8f z = {};
    v8f accP[3];
#pragma unroll
    for (int d = 0; d < 3; ++d) accP[d] = z;
#pragma unroll 2
    for (int k0 = 0; k0 < ci; k0 += 4) {
      const int ich = k0 + half * 2;
      const v2f af = *(const v2f*)(wfrow + ich);  // 8B-aligned vector load
#pragma unroll
      for (int d = 0; d < 3; ++d) {
        v2f bm;
        if (GATHER) {
          bm.x = gfeat(Xb, ich, d, ncol, nbr, c, Cin);
          bm.y = gfeat(Xb, ich + 1, d, ncol, nbr, c, Cin);
        } else {
          bm.x = Xb[(ich * 3 + d) * NPTS + ncol];
          bm.y = Xb[((ich + 1) * 3 + d) * NPTS + ncol];
        }
        accP[d] = wmma4(af, bm, accP[d]);
      }
    }
#pragma unroll
    for (int v = 0; v < 8; ++v) {
      const float nv = sqrtf(accP[0][v] * accP[0][v] + accP[1][v] * accP[1][v] +
                             accP[2][v] * accP[2][v]) + EPSV;
      pn[v] += nv;
      pn2[v] += nv * nv;
    }
  }
  // reduce across each 16-lane half (same output row), then 2 atomics/row
#pragma unroll
  for (int v = 0; v < 8; ++v) {
    float s = pn[v], s2 = pn2[v];
#pragma unroll
    for (int off = 8; off >= 1; off >>= 1) {
      s += __shfl_xor(s, off);
      s2 += __shfl_xor(s2, off);
    }
    if (lcol == 0) {
      const int o = o0 + v + half * 8;
      atomicAdd(&sumN[o], s);
      atomicAdd(&sumN2[o], s2);
    }
  }
}

__global__ void bn_finalize_kernel(const float* __restrict__ sumN,
                                   const float* __restrict__ sumN2, int co,
                                   float invcnt, float* __restrict__ mu,
                                   float* __restrict__ invs) {
  const int o = blockIdx.x * blockDim.x + threadIdx.x;
  if (o < co) {
    const float m = sumN[o] * invcnt;
    const float var = sumN2[o] * invcnt - m * m;
    mu[o] = m;
    invs[o] = rsqrtf(var + BNEPS);
  }
}

// ---------------------------------------------------------------------------
// Pass B (blocks 1-4): recompute p & d GEMMs, apply BN + branchless
// vector-ReLU per neighbor, accumulate mean over k, store next activation.
// ---------------------------------------------------------------------------
__global__ __launch_bounds__(32) void vn_passB_graph(
    const float* __restrict__ Xin, int bstride, int Cin,
    const float* __restrict__ Wf, const float* __restrict__ Wd,
    const int* __restrict__ nidx, int c, int ci, const float* __restrict__ mu,
    const float* __restrict__ invs, const float* __restrict__ gamma,
    const float* __restrict__ beta, float* __restrict__ Xout, int obstride) {
  const int lane = threadIdx.x, half = lane >> 4, lcol = lane & 15;
  const int n0 = blockIdx.x * 16, o0 = blockIdx.y * 16, b = blockIdx.z;
  const float* Xb = Xin + b * bstride;
  const int ncol = n0 + lcol;
  const float* wfrow = Wf + (o0 + lcol) * ci;
  const float* wdrow = Wd + (o0 + lcol) * ci;
  const int* irow = nidx + (b * NPTS + ncol) * KNN;
  float muv[8], isv[8], gv[8], bv[8];
#pragma unroll
  for (int v = 0; v < 8; ++v) {
    const int o = o0 + v + half * 8;
    muv[v] = mu[o]; isv[v] = invs[o]; gv[v] = gamma[o]; bv[v] = beta[o];
  }
  v8f z = {};
  v8f mean[3];
#pragma unroll
  for (int d = 0; d < 3; ++d) mean[d] = z;

  for (int k = 0; k < KNN; ++k) {
    const int nbr = irow[k];
    v8f accP[3], accD[3];
#pragma unroll
    for (int d = 0; d < 3; ++d) { accP[d] = z; accD[d] = z; }
#pragma unroll 2
    for (int k0 = 0; k0 < ci; k0 += 4) {
      const int ich = k0 + half * 2;
      const v2f af = *(const v2f*)(wfrow + ich);
      const v2f ad = *(const v2f*)(wdrow + ich);
#pragma unroll
      for (int d = 0; d < 3; ++d) {
        v2f bm;
        bm.x = gfeat(Xb, ich, d, ncol, nbr, c, Cin);
        bm.y = gfeat(Xb, ich + 1, d, ncol, nbr, c, Cin);
        accP[d] = wmma4(af, bm, accP[d]);
        accD[d] = wmma4(ad, bm, accD[d]);
      }
    }
#pragma unroll
    for (int v = 0; v < 8; ++v) {
      float p0 = accP[0][v], p1 = accP[1][v], p2 = accP[2][v];
      const float d0 = accD[0][v], d1 = accD[1][v], d2 = accD[2][v];
      const float nv = sqrtf(p0 * p0 + p1 * p1 + p2 * p2) + EPSV;
      const float nbn = (nv - muv[v]) * isv[v] * gv[v] + bv[v];
      const float sc = nbn / nv;
      p0 *= sc; p1 *= sc; p2 *= sc;
      const float dot = p0 * d0 + p1 * d1 + p2 * d2;
      // branchless vector-ReLU: cf = min(dot,0)/(|d|^2+eps)
      const float cf = fminf(dot, 0.f) / (d0 * d0 + d1 * d1 + d2 * d2 + EPSV);
      p0 -= cf * d0; p1 -= cf * d1; p2 -= cf * d2;
      mean[0][v] += p0; mean[1][v] += p1; mean[2][v] += p2;
    }
  }
  float* Ob = Xout + b * obstride;
  const float sck = 1.f / (float)KNN;
#pragma unroll
  for (int v = 0; v < 8; ++v) {
    const int o = o0 + v + half * 8;
#pragma unroll
    for (int d = 0; d < 3; ++d)
      Ob[(o * 3 + d) * NPTS + ncol] = mean[d][v] * sck;
  }
}

// ---------------------------------------------------------------------------
// Pass B (block 5): plain GEMM, BN + branchless vector-ReLU, mean over n via
// half-wave shuffle reduction + atomicAdd into d_out (4,1024,3).
// ---------------------------------------------------------------------------
__global__ __launch_bounds__(32) void vn_passB_final(
    const float* __restrict__ Xin, int bstride, const float* __restrict__ Wf,
    const float* __restrict__ Wd, int ci, const float* __restrict__ mu,
    const float* __restrict__ invs, const float* __restrict__ gamma,
    const float* __restrict__ beta, float* __restrict__ out) {
  const int lane = threadIdx.x, half = lane >> 4, lcol = lane & 15;
  const int n0 = blockIdx.x * 16, o0 = blockIdx.y * 16, b = blockIdx.z;
  const float* Xb = Xin + b * bstride;
  const int ncol = n0 + lcol;
  const float* wfrow = Wf + (o0 + lcol) * ci;
  const float* wdrow = Wd + (o0 + lcol) * ci;
  float muv[8], isv[8], gv[8], bv[8];
#pragma unroll
  for (int v = 0; v < 8; ++v) {
    const int o = o0 + v + half * 8;
    muv[v] = mu[o]; isv[v] = invs[o]; gv[v] = gamma[o]; bv[v] = beta[o];
  }
  v8f z = {};
  v8f accP[3], accD[3];
#pragma unroll
  for (int d = 0; d < 3; ++d) { accP[d] = z; accD[d] = z; }
#pragma unroll 2
  for (int k0 = 0; k0 < ci; k0 += 4) {
    const int ich = k0 + half * 2;
    const v2f af = *(const v2f*)(wfrow + ich);
    const v2f ad = *(const v2f*)(wdrow + ich);
#pragma unroll
    for (int d = 0; d < 3; ++d) {
      v2f bm;
      bm.x = Xb[(ich * 3 + d) * NPTS + ncol];
      bm.y = Xb[((ich + 1) * 3 + d) * NPTS + ncol];
      accP[d] = wmma4(af, bm, accP[d]);
      accD[d] = wmma4(ad, bm, accD[d]);
    }
  }
#pragma unroll
  for (int v = 0; v < 8; ++v) {
    float p0 = accP[0][v], p1 = accP[1][v], p2 = accP[2][v];
    const float d0 = accD[0][v], d1 = accD[1][v], d2 = accD[2][v];
    const float nv = sqrtf(p0 * p0 + p1 * p1 + p2 * p2) + EPSV;
    const float nbn = (nv - muv[v]) * isv[v] * gv[v] + bv[v];
    const float sc = nbn / nv;
    p0 *= sc; p1 *= sc; p2 *= sc;
    const float dot = p0 * d0 + p1 * d1 + p2 * d2;
    const float cf = fminf(dot, 0.f) / (d0 * d0 + d1 * d1 + d2 * d2 + EPSV);
    accP[0][v] = p0 - cf * d0;
    accP[1][v] = p1 - cf * d1;
    accP[2][v] = p2 - cf * d2;
  }
  const float scn = 1.f / (float)NPTS;
#pragma unroll
  for (int v = 0; v < 8; ++v) {
    const int o = o0 + v + half * 8;
#pragma unroll
    for (int d = 0; d < 3; ++d) {
      float s = accP[d][v];
#pragma unroll
      for (int off = 8; off >= 1; off >>= 1) s += __shfl_xor(s, off);
      if (lcol == 0) atomicAdd(&out[b * 3072 + o * 3 + d], s * scn);
    }
  }
}

// ---------------------------------------------------------------------------
// Host orchestration (graph-capture safe: only stream launches + MemsetAsync)
// ---------------------------------------------------------------------------
extern "C" void kernel_launch(void* const* d_in, const int* in_sizes, int n_in,
                              void* d_out, int out_size, void* d_ws,
                              size_t ws_size, hipStream_t stream) {
  (void)in_sizes; (void)n_in; (void)out_size; (void)ws_size;
  const float* x = (const float*)d_in[0];
  const float *wF[5], *wD[5], *gm[5], *bt[5];
  for (int i = 0; i < 5; ++i) {
    wF[i] = (const float*)d_in[1 + 4 * i];
    wD[i] = (const float*)d_in[2 + 4 * i];
    gm[i] = (const float*)d_in[3 + 4 * i];
    bt[i] = (const float*)d_in[4 + 4 * i];
  }
  // Workspace layout (floats)
  float* ws = (float*)d_ws;
  float* dist = ws;                            // 4 * 1024 * 1024
  float* xcbuf = dist + 4 * 1024 * 1024;       // 4 * 1536 * 1024 (concat x1..x4)
  float* xx = xcbuf + 4 * 1536 * 1024;         // 4096
  float* sumN = xx + 4096;                     // 1024
  float* sumN2 = sumN + 1024;                  // 1024
  float* mu = sumN2 + 1024;                    // 1024
  float* invs = mu + 1024;                     // 1024
  float* xpad = invs + 1024;                   // 4 * 4 * 1024
  float* w1fp = xpad + 4 * 4 * 1024;           // 64 * 4
  float* w1dp = w1fp + 256;                    // 64 * 4
  int* nidx = (int*)(w1dp + 256);              // 4 * 1024 * 20

  // Pad block-1 operands so all hot loops are guard-free (C, ci % 4 == 0).
  padx_kernel<<<dim3(64), 256, 0, stream>>>(x, xpad);
  padw_kernel<<<dim3(1), 256, 0, stream>>>(wF[0], w1fp);
  padw_kernel<<<dim3(1), 256, 0, stream>>>(wD[0], w1dp);

  const int xcs = 1536 * NPTS;  // per-batch stride of concat buffer
  struct Blk {
    const float* Xin; int bstride; int Cin; int c; int ci; int co;
    const float* Wf; const float* Wd; float* Xout;
  };
  const Blk blk[4] = {
      {xpad, 4 * NPTS, 4, 1, 4, 64, w1fp, w1dp, xcbuf},        // rows   0..191
      {xcbuf, xcs, 192, 64, 128, 64, wF[1], wD[1], xcbuf + 192 * NPTS},
      {xcbuf + 192 * NPTS, xcs, 192, 64, 128, 128, wF[2], wD[2],
       xcbuf + 384 * NPTS},
      {xcbuf + 384 * NPTS, xcs, 384, 128, 256, 256, wF[3], wD[3],
       xcbuf + 768 * NPTS},
  };

  for (int i = 0; i < 4; ++i) {
    sqnorm_kernel<<<dim3((4 * NPTS) / 256), 256, 0, stream>>>(
        blk[i].Xin, blk[i].bstride, blk[i].Cin, xx);
    gram_dist_kernel<<<dim3(64, 64, 4), 32, 0, stream>>>(
        blk[i].Xin, blk[i].bstride, blk[i].Cin, xx, dist);
    topk_kernel<<<dim3((4 * NPTS) / 8), 256, 0, stream>>>(dist, nidx);
    hipMemsetAsync(sumN, 0, 2 * 1024 * sizeof(float), stream);
    vn_passA<true><<<dim3(64, blk[i].co / 16, 4), 32, 0, stream>>>(
        blk[i].Xin, blk[i].bstride, blk[i].Cin, blk[i].Wf, nidx, blk[i].c,
        blk[i].ci, KNN, sumN, sumN2);
    bn_finalize_kernel<<<dim3((blk[i].co + 255) / 256), 256, 0, stream>>>(
        sumN, sumN2, blk[i].co, 1.f / (float)(4 * NPTS * KNN), mu, invs);
    vn_passB_graph<<<dim3(64, blk[i].co / 16, 4), 32, 0, stream>>>(
        blk[i].Xin, blk[i].bstride, blk[i].Cin, blk[i].Wf, blk[i].Wd, nidx,
        blk[i].c, blk[i].ci, mu, invs, gm[i], bt[i], blk[i].Xout, xcs);
  }

  // Block 5: ci=512, co=1024, positions = n, mean over n -> d_out (4,1024,3)
  hipMemsetAsync(sumN, 0, 2 * 1024 * sizeof(float), stream);
  vn_passA<false><<<dim3(64, 1024 / 16, 4), 32, 0, stream>>>(
      xcbuf, xcs, 1536, wF[4], nidx, /*c=*/0, /*ci=*/512, /*knn=*/1, sumN,
      sumN2);
  bn_finalize_kernel<<<dim3(4), 256, 0, stream>>>(
      sumN, sumN2, 1024, 1.f / (float)(4 * NPTS), mu, invs);
  hipMemsetAsync(d_out, 0, 4 * 1024 * 3 * sizeof(float), stream);
  vn_passB_final<<<dim3(64, 1024 / 16, 4), 32, 0, stream>>>(
      xcbuf, xcs, wF[4], wD[4], 512, mu, invs, gm[4], bt[4], (float*)d_out);
}